// Router_37022618091707
// MI455X (gfx1250) — compile-verified
//
#include <hip/hip_runtime.h>
#include <math.h>

namespace {

constexpr int T    = 32768;   // tokens
constexpr int D    = 2048;    // hidden dim
constexpr int E    = 64;      // experts
constexpr int KTOP = 8;       // top-k

constexpr int KC       = 64;           // K-chunk (floats) staged in LDS
constexpr int NCHUNK   = D / KC;       // 32 chunks
constexpr int ROWS     = 32;           // token rows per wave = 2 strips of 16
constexpr int BUF_FLT  = ROWS * KC;    // 2048 floats = 8 KB per buffer
// per wave: 2 buffers (16 KB) -> per block (8 waves): 128 KB dynamic LDS

typedef float v2f  __attribute__((ext_vector_type(2)));
typedef float v8f  __attribute__((ext_vector_type(8)));
typedef unsigned int u32x4 __attribute__((ext_vector_type(4)));
typedef int   i32x4 __attribute__((ext_vector_type(4)));
typedef int   i32x8 __attribute__((ext_vector_type(8)));

#if defined(__has_builtin)
#if __has_builtin(__builtin_amdgcn_tensor_load_to_lds)
#define USE_TDM 1
#endif
#endif

__device__ __forceinline__ float gumbel_of(float x) {
  float uc = fminf(fmaxf(x, 1e-6f), 1.0f - 1e-6f);
  return -__logf(-__logf(uc));
}

// Fill one 32-row x 64-float h chunk into LDS.
#ifdef USE_TDM
// Single TDM descriptor: 2-D tile, 4B elements, row stride D.
__device__ __forceinline__ void fill_chunk(const float* gsrc,
                                           unsigned ldsDst, int /*lane*/) {
  const uint64_t ga = (uint64_t)(uintptr_t)gsrc;

  u32x4 g0;
  g0[0] = 1u;                                   // count=1 (valid user D#)
  g0[1] = ldsDst;                               // lds_addr (bytes)
  g0[2] = (unsigned)(ga & 0xffffffffu);         // global_addr[31:0]
  g0[3] = (unsigned)((ga >> 32) & 0x01ffffffu)  // global_addr[56:32]
        | (2u << 30);                           // type=2 ("image")

  i32x8 g1;
  g1[0] = 0x00020000;          // workgroup_mask=0 | data_size=2 (4B) | no flags
  g1[1] = (KC   & 0xffff) << 16;                // barrier_addr=0 | tensor_dim0 lo
  g1[2] = (KC >> 16) | ((ROWS & 0xffff) << 16); // tensor_dim0 hi | tensor_dim1 lo
  g1[3] = (ROWS >> 16) | (KC << 16);            // tensor_dim1 hi | tile_dim0=64
  g1[4] = ROWS;                                 // tile_dim1=32 | tile_dim2=0
  g1[5] = D;                                    // tensor_dim0_stride[31:0]=2048
  g1[6] = 0;                                    // stride0 hi | stride1 lo
  g1[7] = 0;                                    // tensor_dim1_stride hi

  i32x4 gz4 = {0, 0, 0, 0};                     // groups 2/3: unused (2-D)
  i32x8 gz8 = {0, 0, 0, 0, 0, 0, 0, 0};         // extra group (clang-23 form)
  __builtin_amdgcn_tensor_load_to_lds(g0, g1, gz4, gz4, gz8, 0);
}
__device__ __forceinline__ void wait_fills_steady() {
  asm volatile("s_wait_tensorcnt 0x1" ::: "memory");   // oldest chunk landed
}
__device__ __forceinline__ void wait_fills_all() {
  asm volatile("s_wait_tensorcnt 0x0" ::: "memory");
}
#else
// Fallback: 16x GLOBAL_LOAD_ASYNC_TO_LDS_B128 (instr j moves rows 2j..2j+1).
__device__ __forceinline__ void fill_chunk(const float* gsrc,
                                           unsigned ldsDst, int lane) {
  const int row = lane >> 4;           // 0..1
  const int c16 = (lane & 15) * 16;    // byte column within row
  unsigned la = ldsDst + (unsigned)(row * (KC * 4) + c16);
  uint64_t ga = (uint64_t)(uintptr_t)((const char*)gsrc +
                                      (size_t)row * (D * 4) + c16);
#pragma unroll
  for (int j = 0; j < 16; ++j) {
    asm volatile("global_load_async_to_lds_b128 %0, %1, off"
                 :: "v"(la), "v"(ga)
                 : "memory");
    la += 2u * (KC * 4u);
    ga += 2ull * (D * 4ull);
  }
}
__device__ __forceinline__ void wait_fills_steady() {
  asm volatile("s_wait_asynccnt 0x10" ::: "memory");   // oldest 16 landed
}
__device__ __forceinline__ void wait_fills_all() {
  asm volatile("s_wait_asynccnt 0x0" ::: "memory");
}
#endif

__device__ __forceinline__ v8f wmma4(v2f a, v2f b, v8f c) {
  return __builtin_amdgcn_wmma_f32_16x16x4_f32(false, a, false, b, (short)0, c,
                                               false, false);
}

// Per-strip epilogue: bias, logits/probs stores, gumbel top-8 mask.
//  C layout: vgpr g, lane L -> row M = 8*(L/16)+g, col N = L%16 (tile n -> e=16n+N)
__device__ __forceinline__ void epilogue(v8f acc0, v8f acc1, v8f acc2, v8f acc3,
                                         int row0, int laneLo, int laneHi,
                                         const float* __restrict__ bias,
                                         const float* __restrict__ u,
                                         float* __restrict__ outM,
                                         float* __restrict__ outP,
                                         float* __restrict__ outL) {
  // bias on identity expert (e = 63 -> tile 3, laneLo 15)
  const float b_id = bias[E - 1];
  if (laneLo == 15) {
#pragma unroll
    for (int g = 0; g < 8; ++g) acc3[g] += b_id;
  }

  // logits
#pragma unroll
  for (int g = 0; g < 8; ++g) {
    const size_t r = (size_t)(row0 + 8 * laneHi + g) * E;
    outL[r + 0  + laneLo] = acc0[g];
    outL[r + 16 + laneLo] = acc1[g];
    outL[r + 32 + laneLo] = acc2[g];
    outL[r + 48 + laneLo] = acc3[g];
  }

  // softmax (TEMP == 1): reduce across 16 lanes x 4 tiles
#pragma unroll
  for (int g = 0; g < 8; ++g) {
    float m = fmaxf(fmaxf(acc0[g], acc1[g]), fmaxf(acc2[g], acc3[g]));
#pragma unroll
    for (int s = 1; s < 16; s <<= 1) m = fmaxf(m, __shfl_xor(m, s, 32));
    const float e0 = __expf(acc0[g] - m);
    const float e1 = __expf(acc1[g] - m);
    const float e2 = __expf(acc2[g] - m);
    const float e3 = __expf(acc3[g] - m);
    float ssum = (e0 + e1) + (e2 + e3);
#pragma unroll
    for (int s = 1; s < 16; s <<= 1) ssum += __shfl_xor(ssum, s, 32);
    const float inv = 1.0f / ssum;
    const size_t r = (size_t)(row0 + 8 * laneHi + g) * E;
    outP[r + 0  + laneLo] = e0 * inv;
    outP[r + 16 + laneLo] = e1 * inv;
    outP[r + 32 + laneLo] = e2 * inv;
    outP[r + 48 + laneLo] = e3 * inv;
  }

  // gumbel-perturbed logits
  float z0[8], z1[8], z2[8], z3[8];
#pragma unroll
  for (int g = 0; g < 8; ++g) {
    const size_t r = (size_t)(row0 + 8 * laneHi + g) * E;
    z0[g] = acc0[g] + gumbel_of(u[r + 0  + laneLo]);
    z1[g] = acc1[g] + gumbel_of(u[r + 16 + laneLo]);
    z2[g] = acc2[g] + gumbel_of(u[r + 32 + laneLo]);
    z3[g] = acc3[g] + gumbel_of(u[r + 48 + laneLo]);
  }

  // iterative top-8 per row via cross-lane argmax in 16-lane groups
  unsigned chosen = 0u;  // bit (n*8 + g) => expert 16n+laneLo picked for row g
#pragma unroll
  for (int g = 0; g < 8; ++g) {
    for (int k = 0; k < KTOP; ++k) {
      float bv = z0[g]; int bn = 0;
      if (z1[g] > bv) { bv = z1[g]; bn = 1; }
      if (z2[g] > bv) { bv = z2[g]; bn = 2; }
      if (z3[g] > bv) { bv = z3[g]; bn = 3; }
      int be = bn * 16 + laneLo;
#pragma unroll
      for (int s = 1; s < 16; s <<= 1) {
        const float ov = __shfl_xor(bv, s, 32);
        const int   oe = __shfl_xor(be, s, 32);
        if (ov > bv || (ov == bv && oe < be)) { bv = ov; be = oe; }
      }
      if ((be & 15) == laneLo) {
        const int n = be >> 4;
        chosen |= 1u << (n * 8 + g);
        if      (n == 0) z0[g] = -INFINITY;
        else if (n == 1) z1[g] = -INFINITY;
        else if (n == 2) z2[g] = -INFINITY;
        else             z3[g] = -INFINITY;
      }
    }
  }

  // mask (bool -> float)
#pragma unroll
  for (int g = 0; g < 8; ++g) {
    const size_t r = (size_t)(row0 + 8 * laneHi + g) * E;
    outM[r + 0  + laneLo] = ((chosen >> (0  + g)) & 1u) ? 1.0f : 0.0f;
    outM[r + 16 + laneLo] = ((chosen >> (8  + g)) & 1u) ? 1.0f : 0.0f;
    outM[r + 32 + laneLo] = ((chosen >> (16 + g)) & 1u) ? 1.0f : 0.0f;
    outM[r + 48 + laneLo] = ((chosen >> (24 + g)) & 1u) ? 1.0f : 0.0f;
  }
}

// One wave computes two 16-token strips (32 tokens x 64 experts).
// h chunks stream through per-wave double-buffered LDS via TDM tensor loads
// (fallback: async-to-LDS); W fragments come from global (L2/L0 resident).
__global__ __launch_bounds__(256, 1) void router_wmma_f32(
    const float* __restrict__ h,
    const float* __restrict__ W,
    const float* __restrict__ bias,
    const float* __restrict__ u,
    float* __restrict__ outM,
    float* __restrict__ outP,
    float* __restrict__ outL)
{
  extern __shared__ float lds[];

  const int lane   = threadIdx.x & 31;
  const int wave   = threadIdx.x >> 5;
  const int pair   = blockIdx.x * 8 + wave;   // strip-pair index
  const int row0   = pair * ROWS;             // first of 32 token rows
  const int laneLo = lane & 15;
  const int laneHi = lane >> 4;

  float* wbuf = lds + wave * (2 * BUF_FLT);
  const unsigned wbufAddr = (unsigned)(uintptr_t)wbuf;  // LDS byte address

  const float* hBase = h + (size_t)row0 * D;
  const float* bP    = W + (size_t)laneLo * D + 2 * laneHi;

  // A-fragment feeds from LDS: strip0 rows 0..15, strip1 rows 16..31
  const float* a0P = wbuf + laneLo * KC + 2 * laneHi;
  const float* a1P = wbuf + (16 + laneLo) * KC + 2 * laneHi;

  // prologue: prefetch chunks 0 and 1
  fill_chunk(hBase + 0 * KC, wbufAddr, lane);
  fill_chunk(hBase + 1 * KC, wbufAddr + BUF_FLT * 4, lane);

  v8f a00 = {}, a01 = {}, a02 = {}, a03 = {};   // strip0 accumulators
  v8f a10 = {}, a11 = {}, a12 = {}, a13 = {};   // strip1 accumulators

  for (int c = 0; c < NCHUNK; ++c) {
    if (c + 1 < NCHUNK) wait_fills_steady();    // chunk c has landed
    else                wait_fills_all();

    const float* a0c = a0P + (c & 1) * BUF_FLT;
    const float* a1c = a1P + (c & 1) * BUF_FLT;
    const float* bPc = bP + c * KC;

#pragma unroll 4
    for (int k = 0; k < KC; k += 4) {
      v2f a0 = *(const v2f*)(a0c + k);
      v2f a1 = *(const v2f*)(a1c + k);
      v2f b0 = *(const v2f*)(bPc + k);
      v2f b1 = *(const v2f*)(bPc + k + 16 * D);
      v2f b2 = *(const v2f*)(bPc + k + 32 * D);
      v2f b3 = *(const v2f*)(bPc + k + 48 * D);
      a00 = wmma4(a0, b0, a00);
      a10 = wmma4(a1, b0, a10);
      a01 = wmma4(a0, b1, a01);
      a11 = wmma4(a1, b1, a11);
      a02 = wmma4(a0, b2, a02);
      a12 = wmma4(a1, b2, a12);
      a03 = wmma4(a0, b3, a03);
      a13 = wmma4(a1, b3, a13);
    }

    if (c + 2 < NCHUNK) {
      // all ds reads of the buffer being recycled have retired
      asm volatile("s_wait_dscnt 0x0" ::: "memory");
      fill_chunk(hBase + (c + 2) * KC,
                 wbufAddr + (c & 1) * (BUF_FLT * 4), lane);
    }
  }

  epilogue(a00, a01, a02, a03, row0,      laneLo, laneHi, bias, u, outM, outP, outL);
  epilogue(a10, a11, a12, a13, row0 + 16, laneLo, laneHi, bias, u, outM, outP, outL);
}

} // namespace

extern "C" void kernel_launch(void* const* d_in, const int* in_sizes, int n_in,
                              void* d_out, int out_size, void* d_ws, size_t ws_size,
                              hipStream_t stream) {
  const float* h    = (const float*)d_in[0];
  const float* W    = (const float*)d_in[1];
  const float* bias = (const float*)d_in[2];
  const float* u    = (const float*)d_in[3];

  float* out  = (float*)d_out;
  float* outM = out;                       // mask  [T*E]
  float* outP = out + (size_t)T * E;       // probs [T*E]
  float* outL = out + 2 * (size_t)T * E;   // logits[T*E]

  // 2048 strips / 2 per wave / 8 waves per block = 128 blocks
  dim3 grid(T / ROWS / 8);
  dim3 block(256);
  size_t shmem = 8 * 2 * BUF_FLT * sizeof(float);  // 128 KB
  hipLaunchKernelGGL(router_wmma_f32, grid, block, shmem, stream,
                     h, W, bias, u, outM, outP, outL);
}